// DegreePrediction_36240934044287
// MI455X (gfx1250) — compile-verified
//
#include <hip/hip_runtime.h>

// DegreePrediction on MI455X (gfx1250):
//   - 4096 workgroups x 1 wave32; each WG owns one 64x64 matrix W_p in LDS.
//   - Power iteration with one 64x64 f32 matvec per step via V_WMMA_F32_16X16X4_F32
//     (vector broadcast across all 16 B columns; 4 row tiles x 16 K-steps = 64 wmma).
//   - Deterministic final combine: per-matrix contributions to d_ws, then a
//     fixed-order tree reduction kernel.

#define NN 64
#define PP (NN * NN) /* 4096 matrices */
#define MAX_ITERS 60
#define TOLF 1e-4f

typedef __attribute__((ext_vector_type(2))) float v2f;
typedef __attribute__((ext_vector_type(8))) float v8f;

__device__ __forceinline__ float wave_reduce_add(float v) {
#pragma unroll
  for (int off = 16; off > 0; off >>= 1) v += __shfl_xor(v, off, 32);
  return v;
}

__launch_bounds__(32)
__global__ void power_iter_kernel(const float* __restrict__ x,
                                  const float* __restrict__ r_zeros,
                                  const float* __restrict__ r_const,
                                  const float* __restrict__ weights_t,
                                  const float* __restrict__ weights_r,
                                  float* __restrict__ contrib /* [64][4096] */) {
  __shared__ float As[NN * NN]; // 16 KB matrix
  __shared__ float buf0[NN];
  __shared__ float buf1[NN];

  const int p = blockIdx.x;      // p = s*64 + t
  const int lane = threadIdx.x;  // 0..31 (wave32)
  const int half = lane >> 4;    // 0 or 1 (lane half -> K offset {0,2})
  const int lrow = lane & 15;    // row within 16-row tile

  // ---- Build W_p = weights_r*r_zeros + r_const into LDS (coalesced float4) ----
  const size_t base = (size_t)p * (NN * NN);
  const float4* wr4 = (const float4*)(weights_r + base);
  const float4* rz4 = (const float4*)(r_zeros + base);
  const float4* rc4 = (const float4*)(r_const + base);
  float4* As4 = (float4*)As;
#pragma unroll 4
  for (int i = lane; i < (NN * NN) / 4; i += 32) {
    float4 a = wr4[i], b = rz4[i], c = rc4[i];
    float4 w;
    w.x = fmaf(a.x, b.x, c.x);
    w.y = fmaf(a.y, b.y, c.y);
    w.z = fmaf(a.z, b.z, c.z);
    w.w = fmaf(a.w, b.w, c.w);
    As4[i] = w;
  }
  // v0 = ones/sqrt(64) = 0.125
  buf0[lane] = 0.125f;
  buf0[lane + 32] = 0.125f;
  __syncthreads();

  // ---- matvec: out = As * in (both 64-float LDS vectors) via f32 WMMA ----
  auto matvec = [&](const float* in, float* out) {
    v8f acc[4] = {};
#pragma unroll
    for (int kk = 0; kk < 16; ++kk) {
      const int kbase = kk * 4 + half * 2;
      // B fragment: broadcast in[] across all 16 columns.
      // VGPR0 holds K = kbase, VGPR1 holds K = kbase+1 (lane halves give +0/+2).
      v2f b;
      b.x = in[kbase];
      b.y = in[kbase + 1];
#pragma unroll
      for (int m = 0; m < 4; ++m) {
        const float* ap = As + (m * 16 + lrow) * NN + kbase;
        v2f a;
        a.x = ap[0]; // A[16m+lrow][kbase]   -> VGPR0
        a.y = ap[1]; // A[16m+lrow][kbase+1] -> VGPR1
        acc[m] = __builtin_amdgcn_wmma_f32_16x16x4_f32(
            false, a, false, b, (short)0, acc[m], false, false);
      }
    }
    // All columns of each D tile are identical (= y segment). Extract once:
    // VGPR r, lane half h holds y[16m + 8h + r]; lanes 0 and 16 write it out.
#pragma unroll
    for (int m = 0; m < 4; ++m)
#pragma unroll
      for (int r = 0; r < 8; ++r)
        if (lrow == 0) out[m * 16 + half * 8 + r] = acc[m][r];
    __syncthreads();
  };

  float* ucur = buf0;
  float* tcur = buf1;

  // u = A v0 ; ev0 = v0 . u = sum(u)/8
  matvec(ucur, tcur);
  float ev = wave_reduce_add(tcur[lane] + tcur[lane + 32]) * 0.125f;
  {
    float* tmp = ucur; ucur = tcur; tcur = tmp;
  }

  // Power iteration: one matvec per step.
  //   n2 = |u|^2 ; t = A u ; ev_new = (u.t)/n2 ; v_new = u/|u| ; u <- t/|u|
  float vlo = 0.f, vhi = 0.f, rinv = 1.f;
  for (int it = 0; it < MAX_ITERS; ++it) {
    float ulo = ucur[lane], uhi = ucur[lane + 32];
    float n2 = wave_reduce_add(ulo * ulo + uhi * uhi);
    matvec(ucur, tcur);
    float d = wave_reduce_add(ulo * tcur[lane] + uhi * tcur[lane + 32]);
    float ev_new = d / n2;
    rinv = 1.0f / sqrtf(n2);
    if (fabsf(ev - ev_new) < TOLF || it == MAX_ITERS - 1) {
      vlo = ulo * rinv; // v = freshly computed v_new, as in the reference
      vhi = uhi * rinv;
      break;
    }
    ev = ev_new;
    tcur[lane] *= rinv; // u_next = t / |u|
    tcur[lane + 32] *= rinv;
    float* tmp = ucur; ucur = tcur; tcur = tmp;
    __syncthreads();
  }

  // scale = x[s,t]*wt[s,t]*r_const[s,t,s,s] / v[s]
  const int s = p >> 6;
  const float denom = ucur[s] * rinv;
  const float tval = x[p] * weights_t[p] * r_const[base + (size_t)s * NN + s];
  const float scale = tval / denom;

  contrib[(size_t)lane * PP + p] = scale * vlo;
  contrib[(size_t)(lane + 32) * PP + p] = scale * vhi;
}

__launch_bounds__(128)
__global__ void reduce_kernel(const float* __restrict__ contrib,
                              float* __restrict__ out) {
  __shared__ float red[128];
  const int n = blockIdx.x;
  const int tid = threadIdx.x;
  const float* row = contrib + (size_t)n * PP;
  float s = 0.f;
  for (int q = tid; q < PP; q += 128) s += row[q]; // fixed order per thread
  red[tid] = s;
  __syncthreads();
#pragma unroll
  for (int w = 64; w > 0; w >>= 1) {
    if (tid < w) red[tid] += red[tid + w];
    __syncthreads();
  }
  if (tid == 0) out[n] = red[0];
}

extern "C" void kernel_launch(void* const* d_in, const int* in_sizes, int n_in,
                              void* d_out, int out_size, void* d_ws,
                              size_t ws_size, hipStream_t stream) {
  const float* x = (const float*)d_in[0];
  const float* r_zeros = (const float*)d_in[1];
  const float* r_const = (const float*)d_in[2];
  const float* weights_t = (const float*)d_in[3];
  const float* weights_r = (const float*)d_in[4];
  float* contrib = (float*)d_ws; // 64*4096*4 = 1 MiB scratch

  power_iter_kernel<<<PP, 32, 0, stream>>>(x, r_zeros, r_const, weights_t,
                                           weights_r, contrib);
  reduce_kernel<<<NN, 128, 0, stream>>>(contrib, (float*)d_out);
}